// DWTFrontEnd_1984274891240
// MI455X (gfx1250) — compile-verified
//
#include <hip/hip_runtime.h>
#include <stdint.h>

typedef float        v2f __attribute__((ext_vector_type(2)));
typedef float        v8f __attribute__((ext_vector_type(8)));
typedef unsigned int v4u __attribute__((ext_vector_type(4)));
typedef int          v4i __attribute__((ext_vector_type(4)));
typedef int          v8i __attribute__((ext_vector_type(8)));

#define N_FULL 8192
#define L1 4099
#define L2 2053
#define L3 1030
#define NT 256
#define NW 8   // waves per block (wave32)

// 8-tap filters (sym4-style) from the reference
__constant__ float kRecLo[8] = {
    0.23037781330885523f,  0.7148465705525415f,   0.6308807679295904f,
   -0.02798376941698385f, -0.18703481171888114f,  0.030841381835986965f,
    0.032883011666982945f,-0.010597401784997278f };
__constant__ float kRecHi[8] = {
   -0.010597401784997278f,-0.032883011666982945f, 0.030841381835986965f,
    0.18703481171888114f, -0.02798376941698385f, -0.6308807679295904f,
    0.7148465705525415f,  -0.23037781330885523f };
__constant__ float kDecLo[8] = {
   -0.010597401784997278f, 0.032883011666982945f, 0.030841381835986965f,
   -0.18703481171888114f, -0.02798376941698385f,  0.6308807679295904f,
    0.7148465705525415f,   0.23037781330885523f };
__constant__ float kDecHi[8] = {
   -0.23037781330885523f,  0.7148465705525415f,  -0.6308807679295904f,
   -0.02798376941698385f,  0.18703481171888114f,  0.030841381835986965f,
   -0.032883011666982945f,-0.010597401784997278f };

// Banded synthesis matrix entry: y[16T+p] = sum_q A[p][q] * v[8T+q], q in [0,12)
// A[p][q] = F[2s+pp] where pp=(p+1)&1, s = q - (p-1+pp)/2, 0<=s<4; else 0.
__device__ __forceinline__ float aval(int p, int q, const float* F) {
  const int pp = (p + 1) & 1;
  const int s  = q - ((p - 1 + pp) >> 1);
  return (s >= 0 && s < 4) ? F[2 * s + pp] : 0.0f;
}

// One upsample(x2, zero-pad 1)+8-tap filter stage (transposed conv), Ly = 2*Lv - 6.
// WMMA f32 16x16x4 over 16-output-phase x 16-tile blocks, K=12; scalar tail.
__device__ __forceinline__ void u_stage(const float* __restrict__ v, int Lv,
                                        float* __restrict__ y, int Ly,
                                        const float* __restrict__ F) {
  const int tid  = threadIdx.x;
  const int lane = tid & 31;
  const int wave = tid >> 5;
  const int p    = lane & 15;
  const int kh   = (lane >> 4) << 1;   // K-pair base within a K=4 chunk

  // A fragments (16x4 f32: lanes 0-15 hold K={0,1}, lanes 16-31 hold K={2,3})
  v2f a0, a1, a2;
  a0.x = aval(p, 0 + kh, F);  a0.y = aval(p, 1 + kh, F);
  a1.x = aval(p, 4 + kh, F);  a1.y = aval(p, 5 + kh, F);
  a2.x = aval(p, 8 + kh, F);  a2.y = aval(p, 9 + kh, F);

  const int nFull = Ly >> 8;           // 256 outputs per 16x16 block
  for (int g = wave; g < nFull; g += NW) {
    const int T0 = g << 4;
    const int t  = lane & 15;          // tile (N) index within block
    const int vb = ((T0 + t) << 3) + kh;
    v2f b0, b1, b2;                    // B 4x16: lanes 0-15 K={0,1}, 16-31 K={2,3}
    b0.x = v[vb + 0]; b0.y = v[vb + 1];
    b1.x = v[vb + 4]; b1.y = v[vb + 5];
    b2.x = v[vb + 8]; b2.y = v[vb + 9];
    v8f acc = {};
    acc = __builtin_amdgcn_wmma_f32_16x16x4_f32(false, a0, false, b0, (short)0, acc, false, false);
    acc = __builtin_amdgcn_wmma_f32_16x16x4_f32(false, a1, false, b1, (short)0, acc, false, false);
    acc = __builtin_amdgcn_wmma_f32_16x16x4_f32(false, a2, false, b2, (short)0, acc, false, false);
    // C layout: lanes 0-15 M=r (N=lane), lanes 16-31 M=r+8 -> 8 consecutive floats/lane
    float* dst = y + ((T0 + t) << 4) + ((lane >> 4) << 3);
    *(float4*)(dst + 0) = make_float4(acc[0], acc[1], acc[2], acc[3]);
    *(float4*)(dst + 4) = make_float4(acc[4], acc[5], acc[6], acc[7]);
  }
  // scalar tail (Ly not a multiple of 256)
  for (int j = (nFull << 8) + tid; j < Ly; j += NT) {
    const int pp   = (j + 1) & 1;
    const int base = (j - 1 + pp) >> 1;
    float s = 0.0f;
#pragma unroll
    for (int k = 0; k < 4; ++k) {
      const int vi = base + k;
      const float vv = (vi >= 0 && vi < Lv) ? v[vi] : 0.0f;
      s = fmaf(F[2 * k + pp], vv, s);
    }
    y[j] = s;
  }
}

// One analysis level: symmetric-extend (6 left / 7 right), 8-tap, stride 2.
__device__ __forceinline__ void dwt_level(const float* __restrict__ s, int L,
                                          float* __restrict__ dA,
                                          float* __restrict__ dD, int Lo) {
  float hl[8], hh[8];
#pragma unroll
  for (int t = 0; t < 8; ++t) { hl[t] = kRecLo[t]; hh[t] = kRecHi[t]; }
  for (int m = threadIdx.x; m < Lo; m += NT) {
    float a = 0.0f, d = 0.0f;
    const int i0 = m << 1;
#pragma unroll
    for (int t = 0; t < 8; ++t) {
      const int i  = i0 + t;
      const int si = (i < 6) ? (5 - i) : ((i < L + 6) ? (i - 6) : (2 * L + 5 - i));
      const float xv = s[si];
      a = fmaf(hl[t], xv, a);
      d = fmaf(hh[t], xv, d);
    }
    dA[m] = a; dD[m] = d;
  }
}

__global__ __launch_bounds__(NT)
void wavelet_bands_kernel(const float* __restrict__ xin,
                          float* __restrict__ out, int rows) {
  __shared__ __align__(16) float xbuf[N_FULL + 16];       // row, then synthesis ping buffer
  __shared__ __align__(16) float c1A[L1 + 17], c1D[L1 + 17];
  __shared__ __align__(16) float c2A[L2 + 19], c2D[L2 + 19];
  __shared__ __align__(16) float c3A[L3 + 18], c3D[L3 + 18];
  __shared__ __align__(16) float sA[2054 + 18];           // level-3 synthesis output

  const int tid = threadIdx.x;
  const int row = blockIdx.x;

  if (tid < 16) {  // zero pads so WMMA over-reads hit zeros
    c1A[L1 + tid] = 0.0f; c1D[L1 + tid] = 0.0f;
    c2A[L2 + tid] = 0.0f; c2D[L2 + tid] = 0.0f;
    c3A[L3 + tid] = 0.0f; c3D[L3 + tid] = 0.0f;
    sA[2054 + tid] = 0.0f; xbuf[N_FULL + tid] = 0.0f;
  }

  // --- TDM: DMA the whole 32KB row into LDS (wave 0 only), wait on TENSORcnt ---
  if (tid < 32) {
    const uint64_t gaddr = (uint64_t)(uintptr_t)(xin + (size_t)row * N_FULL);
    const unsigned ldsa  = (unsigned)(uintptr_t)&xbuf[0];  // low 32 bits = LDS byte offset
    v4u g0 = {};
    g0.x = 1u;                                             // count=1
    g0.y = ldsa;                                           // lds_addr
    g0.z = (unsigned)(gaddr & 0xffffffffull);              // global_addr[31:0]
    g0.w = (unsigned)((gaddr >> 32) & 0x1ffffffull) | (2u << 30);  // addr[56:32] | type=2
    v8i g1 = {};
    g1[0] = (int)(2u << 16);                               // data_size = 4B
    g1[1] = (int)(((unsigned)N_FULL & 0xffffu) << 16);     // tensor_dim0[15:0]
    g1[2] = (int)((((unsigned)N_FULL) >> 16) | (1u << 16));// tensor_dim0[31:16] | tensor_dim1=1
    g1[3] = (int)(((unsigned)N_FULL & 0xffffu) << 16);     // tile_dim0 = 8192
    g1[4] = 1;                                             // tile_dim1 = 1
    g1[5] = N_FULL;                                        // tensor_dim0_stride
    g1[6] = 0; g1[7] = 0;
    v4i gz = {};
#if defined(__clang_major__) && (__clang_major__ >= 23)
    v8i gz8 = {};
    __builtin_amdgcn_tensor_load_to_lds(g0, g1, gz, gz, gz8, 0);
#else
    __builtin_amdgcn_tensor_load_to_lds(g0, g1, gz, gz, 0);
#endif
    __builtin_amdgcn_s_wait_tensorcnt(0);
  }
  __syncthreads();

  // --- Analysis: 3 DWT levels (VALU, cheap) ---
  dwt_level(xbuf, N_FULL, c1A, c1D, L1); __syncthreads();
  dwt_level(c1A,  L1,     c2A, c2D, L2); __syncthreads();
  dwt_level(c2A,  L2,     c3A, c3D, L3); __syncthreads();

  const size_t bandStride = (size_t)rows * (size_t)N_FULL;
  float* o0 = out + (size_t)row * N_FULL;

  // --- Synthesis: each stage is a WMMA-based banded GEMM (K=12) ---
  // band 0: keep a3   -> ULO,ULO,ULO
  u_stage(c3A, L3,   sA,   2054, kDecLo); __syncthreads();
  u_stage(sA,  2053, xbuf, 4100, kDecLo); __syncthreads();
  u_stage(xbuf,4099, o0 + 0 * bandStride, N_FULL, kDecLo); __syncthreads();
  // band 1: keep d3   -> UHI,ULO,ULO
  u_stage(c3D, L3,   sA,   2054, kDecHi); __syncthreads();
  u_stage(sA,  2053, xbuf, 4100, kDecLo); __syncthreads();
  u_stage(xbuf,4099, o0 + 1 * bandStride, N_FULL, kDecLo); __syncthreads();
  // band 2: keep d2   -> UHI,ULO
  u_stage(c2D, L2,   xbuf, 4100, kDecHi); __syncthreads();
  u_stage(xbuf,4099, o0 + 2 * bandStride, N_FULL, kDecLo); __syncthreads();
  // band 3: keep d1   -> UHI
  u_stage(c1D, L1,   o0 + 3 * bandStride, N_FULL, kDecHi);
}

extern "C" void kernel_launch(void* const* d_in, const int* in_sizes, int n_in,
                              void* d_out, int out_size, void* d_ws, size_t ws_size,
                              hipStream_t stream) {
  (void)n_in; (void)out_size; (void)d_ws; (void)ws_size;
  const float* x = (const float*)d_in[0];
  float* out = (float*)d_out;
  const int rows = in_sizes[0] / N_FULL;   // 32*64 = 2048
  hipLaunchKernelGGL(wavelet_bands_kernel, dim3(rows), dim3(NT), 0, stream,
                     x, out, rows);
}